// Cluster_51410758533482
// MI455X (gfx1250) — compile-verified
//
#include <hip/hip_runtime.h>

// Problem constants (from reference): x[B=256, C=2048, H=24, W=12], clusters[K=6, C]
#define BATCH 256
#define CH    2048
#define NPOS  288      // H*W = 24*12
#define NK    6

typedef __attribute__((ext_vector_type(2))) float v2f;
typedef __attribute__((ext_vector_type(8))) float v8f;
typedef __attribute__((ext_vector_type(2))) int   v2i;

// One workgroup per batch. 8 waves (wave32). All WMMA in wave-uniform control flow.
__global__ __launch_bounds__(256)
void cluster_means_kernel(const float* __restrict__ x,
                          const float* __restrict__ clusters,
                          float* __restrict__ out) {
  __shared__ float sClus[NK * CH];    // 48 KB: clusters row-major [k][c]
  __shared__ float sZero[4];          // zero pad for padded k-lanes (k>=6)
  __shared__ float sDot[8][16 * 16];  // per-wave dot tile scratch [n][k]
  __shared__ int   sAssign[NPOS];     // per-position cluster assignment
  __shared__ float sCnorm[NK];        // ||c_k||^2
  __shared__ float sInv[NK];          // 1/count_k (0 if empty)

  const int tid  = threadIdx.x;
  const int lane = tid & 31;
  const int wave = tid >> 5;
  const int h    = lane >> 4;   // half-wave id
  const int lm   = lane & 15;
  const int b    = blockIdx.x;

  const float* xb = x + (size_t)b * (CH * NPOS);

  // ---- stage clusters into LDS, zero the pad ----
  for (int i = tid; i < NK * CH; i += 256) sClus[i] = clusters[i];
  if (tid < 4) sZero[tid] = 0.0f;
  __syncthreads();

  // ---- ||c_k||^2 (once per workgroup, negligible) ----
  if (tid < NK) {
    float s = 0.0f;
    const float* row = sClus + tid * CH;
    for (int c = 0; c < CH; ++c) { float v = row[c]; s += v * v; }
    sCnorm[tid] = s;
  }
  __syncthreads();

  // ---- phase 1: dot[n,k] = xf[n,:] . clusters[k,:] via V_WMMA_F32_16X16X4_F32,
  //      argmin_k (cnorm[k] - 2*dot) -> sAssign.  (x_norm is constant over k.)
  for (int t = wave; t < NPOS / 16; t += 8) {
    const int n0 = t * 16;
    // A (16 n-rows x 4 c): lane layout: lanes0-15 K=0(+VGPR j), lanes16-31 K=2(+j)
    const float* pa0 = xb + (size_t)(2 * h + 0) * NPOS + n0 + lm;
    const float* pa1 = xb + (size_t)(2 * h + 1) * NPOS + n0 + lm;
    // B (4 c-rows x 16 k-cols): VGPR j, half h -> row = j + 2h, col = lm
    const float* pb  = (lm < NK) ? (sClus + lm * CH + 2 * h) : sZero;
    const int bstep  = (lm < NK) ? 4 : 0;

    v8f acc = {};
    #pragma unroll 8
    for (int cc = 0; cc < CH / 4; ++cc) {
      v2f a;
      a.x = pa0[(size_t)cc * 4 * NPOS];
      a.y = pa1[(size_t)cc * 4 * NPOS];
      v2f bb = *(const v2f*)(pb + cc * bstep);   // ds_load_b64
      acc = __builtin_amdgcn_wmma_f32_16x16x4_f32(false, a, false, bb,
                                                  (short)0, acc, false, false);
    }
    // D layout: VGPR v, lanes0-15: (n=v, k=lm); lanes16-31: (n=v+8, k=lm)
    float* dw = sDot[wave];
    #pragma unroll
    for (int v = 0; v < 8; ++v) dw[(v + 8 * h) * 16 + lm] = acc[v];
    asm volatile("s_wait_dscnt 0" ::: "memory");
    if (lane < 16) {
      float best = 3.4028235e38f; int bk = 0;
      #pragma unroll
      for (int k = 0; k < NK; ++k) {
        float d = sCnorm[k] - 2.0f * dw[lane * 16 + k];
        if (d < best) { best = d; bk = k; }   // strict '<' == first-min tiebreak
      }
      sAssign[n0 + lane] = bk;
    }
  }
  __syncthreads();

  // ---- counts and reciprocal means scale ----
  if (tid < NK) {
    int cnt = 0;
    for (int n = 0; n < NPOS; ++n) cnt += (sAssign[n] == tid) ? 1 : 0;
    sInv[tid] = (cnt > 0) ? (1.0f / (float)cnt) : 0.0f;   // empty cluster -> 0
  }
  __syncthreads();

  // ---- phase 2: sums[k,c] = onehot^T @ xf via WMMA; x slice is L2-hot now ----
  for (int ct = wave; ct < CH / 16; ct += 8) {
    const int c0 = ct * 16;
    // B (4 n-rows x 16 c-cols): value = x[b, c0+lm, nn + j + 2h]; j-pair contiguous
    const float* pbx = xb + (size_t)(c0 + lm) * NPOS + 2 * h;
    const int*   pa  = sAssign + 2 * h;
    v8f acc = {};
    #pragma unroll 4
    for (int i = 0; i < NPOS / 4; ++i) {
      v2i av = *(const v2i*)(pa + i * 4);      // ds_load_b64 of 2 assignments
      v2f af;                                  // A = onehot: (k=lm) rows
      af.x = (av.x == lm) ? 1.0f : 0.0f;
      af.y = (av.y == lm) ? 1.0f : 0.0f;
      v2f bb = *(const v2f*)(pbx + i * 4);     // global b64, 8B-aligned
      acc = __builtin_amdgcn_wmma_f32_16x16x4_f32(false, af, false, bb,
                                                  (short)0, acc, false, false);
    }
    // D: lanes0-15 hold (k=v, c=c0+lm); lanes16-31 hold k>=8 (padding) -> drop
    if (h == 0) {
      float* ob = out + ((size_t)b * CH + c0 + lm) * NK;   // out[b][c][k]
      #pragma unroll
      for (int v = 0; v < NK; ++v) ob[v] = acc[v] * sInv[v];
    }
  }
}

extern "C" void kernel_launch(void* const* d_in, const int* in_sizes, int n_in,
                              void* d_out, int out_size, void* d_ws, size_t ws_size,
                              hipStream_t stream) {
  const float* x        = (const float*)d_in[0];
  const float* clusters = (const float*)d_in[1];
  float* out            = (float*)d_out;
  cluster_means_kernel<<<dim3(BATCH), dim3(256), 0, stream>>>(x, clusters, out);
}